// MRHG2D_11269994184712
// MI455X (gfx1250) — compile-verified
//
#include <hip/hip_runtime.h>

typedef float v4f __attribute__((ext_vector_type(4)));
typedef float v2f __attribute__((ext_vector_type(2)));
typedef int   v2i __attribute__((ext_vector_type(2)));

#define LVLS   6
#define TBITS  19
#define TSIZE  (1 << TBITS)
#define HMASK  (TSIZE - 1)
#define PRIME_Y 2654435761u
#define LN_EPS 1e-5f
#define BLK    256

#if __has_builtin(__builtin_amdgcn_global_load_async_to_lds_b64) && \
    __has_builtin(__builtin_amdgcn_s_wait_asynccnt)
#define USE_ASYNC_LDS 1
#else
#define USE_ASYNC_LDS 0
#endif

__launch_bounds__(BLK, 4)
__global__ void mrhg2d_kernel(const v2f* __restrict__ pos,
                              const v4f* __restrict__ tables,
                              const float* __restrict__ lw,
                              v4f* __restrict__ out,
                              int n)
{
    const int tid = threadIdx.x;
    const int gi  = blockIdx.x * BLK + tid;
    if (gi >= n) return;

    // ---- Stage this block's positions through LDS with the gfx1250 async
    // ---- LDS-DMA path (ASYNCcnt); fall back to a plain NT load otherwise.
    __shared__ v2f sPos[BLK];
#if USE_ASYNC_LDS
    __builtin_amdgcn_global_load_async_to_lds_b64(
        (__attribute__((address_space(1))) v2i*)(pos + gi),
        (__attribute__((address_space(3))) v2i*)(&sPos[tid]),
        /*offset=*/0, /*cpol=*/0);
    __builtin_amdgcn_s_wait_asynccnt(0);
#else
    sPos[tid] = __builtin_nontemporal_load(pos + gi);
#endif
    const v2f p  = sPos[tid];
    const float px = p.x, py = p.y;

    v4f* __restrict__ outp = out + (size_t)gi * LVLS;

#pragma unroll
    for (int l = 0; l < LVLS; ++l) {
        // cell sizes are exactly 1,2,4,8,16,32 -> multiply by exact reciprocal
        const float invc = 1.0f / (float)(1 << l);
        const float gx = px * invc;
        const float gy = py * invc;
        const float fx = floorf(gx);
        const float fy = floorf(gy);
        const float wx = gx - fx;
        const float wy = gy - fy;
        const int ix0 = (int)fx;
        const int iy0 = (int)fy;

        const unsigned hy0 = (unsigned)iy0 * PRIME_Y;
        const unsigned hy1 = (unsigned)(iy0 + 1) * PRIME_Y;
        const unsigned h00 = ((unsigned)ix0       ^ hy0) & HMASK;
        const unsigned h10 = ((unsigned)(ix0 + 1) ^ hy0) & HMASK;
        const unsigned h01 = ((unsigned)ix0       ^ hy1) & HMASK;
        const unsigned h11 = ((unsigned)(ix0 + 1) ^ hy1) & HMASK;

        // 16B per-lane gathers (global_load_b128); tables stay L2-resident
        // (48 MB total vs 192 MB L2) because the output stream is NT.
        const v4f* __restrict__ tab = tables + (size_t)l * TSIZE;
        const v4f f00 = tab[h00];
        const v4f f10 = tab[h10];
        const v4f f01 = tab[h01];
        const v4f f11 = tab[h11];

        const float w00 = (1.0f - wx) * (1.0f - wy);
        const float w10 = wx * (1.0f - wy);
        const float w01 = (1.0f - wx) * wy;
        const float w11 = wx * wy;

        v4f f = w00 * f00 + w10 * f10 + w01 * f01 + w11 * f11;

        // layer norm over DIM=4
        const float mu = (f.x + f.y + f.z + f.w) * 0.25f;
        v4f d = f - mu;
        const float var = (d.x * d.x + d.y * d.y + d.z * d.z + d.w * d.w) * 0.25f;
        const float s = lw[l] * rsqrtf(var + LN_EPS);
        d = d * s;

        // streaming NT store: do not pollute L2 with the 192MB output
        __builtin_nontemporal_store(d, outp + l);
    }
}

extern "C" void kernel_launch(void* const* d_in, const int* in_sizes, int n_in,
                              void* d_out, int out_size, void* d_ws, size_t ws_size,
                              hipStream_t stream)
{
    const v2f*   pos    = (const v2f*)d_in[0];    // [N,2] f32
    const v4f*   tables = (const v4f*)d_in[1];    // [L,T,4] f32
    const float* lw     = (const float*)d_in[2];  // [L] f32
    v4f*         out    = (v4f*)d_out;            // [N, L*4] f32 -> N*L v4f

    const int n = in_sizes[0] / 2;                // number of points
    const int grid = (n + BLK - 1) / BLK;
    mrhg2d_kernel<<<grid, BLK, 0, stream>>>(pos, tables, lw, out, n);
}